// PerceptaFullSequenceModel_16441134809183
// MI455X (gfx1250) — compile-verified
//
#include <hip/hip_runtime.h>
#include <hip/hip_bf16.h>
#include <math.h>

typedef __attribute__((ext_vector_type(2))) float v2f;
typedef __attribute__((ext_vector_type(8))) float v8f;

#define DIM 36
#define JCHUNK 512          // columns of K staged in LDS per chunk (4 KB)
#define ROWS_PER_BLOCK 128  // 8 waves * 16 rows

// ---------------------------------------------------------------------------
// Kernel 1: project embeddings through all 7 weight matrices (11 output dims).
// Workspace layout (floats):
//   [0   , 2T) Q_prog interleaved   [2T , 4T) K_prog interleaved
//   [4T  , 5T) V_op                 [5T , 6T) V_arg
//   [6T  , 8T) Q_stack interleaved  [8T ,10T) K_stack interleaved
//   [10T ,11T) V_stack
// ---------------------------------------------------------------------------
__global__ __launch_bounds__(256) void proj_kernel(
    const float* __restrict__ E,
    const float* __restrict__ WQp, const float* __restrict__ WKp,
    const float* __restrict__ WVop, const float* __restrict__ WVarg,
    const float* __restrict__ WQs, const float* __restrict__ WKs,
    const float* __restrict__ WVs,
    float* __restrict__ ws, int T)
{
    int t = blockIdx.x * 256 + threadIdx.x;
    if (t >= T) return;
    const float* e = E + (long)t * DIM;

    float qp0 = 0.f, qp1 = 0.f, kp0 = 0.f, kp1 = 0.f;
    float vop = 0.f, varg = 0.f;
    float qs0 = 0.f, qs1 = 0.f, ks0 = 0.f, ks1 = 0.f, vs = 0.f;

#pragma unroll 4
    for (int d = 0; d < DIM; ++d) {
        float x = e[d];
        qp0 = fmaf(x, WQp[d], qp0);       qp1 = fmaf(x, WQp[DIM + d], qp1);
        kp0 = fmaf(x, WKp[d], kp0);       kp1 = fmaf(x, WKp[DIM + d], kp1);
        vop = fmaf(x, WVop[d], vop);      varg = fmaf(x, WVarg[d], varg);
        qs0 = fmaf(x, WQs[d], qs0);       qs1 = fmaf(x, WQs[DIM + d], qs1);
        ks0 = fmaf(x, WKs[d], ks0);       ks1 = fmaf(x, WKs[DIM + d], ks1);
        vs  = fmaf(x, WVs[d], vs);
    }

    ws[2 * t]         = qp0;  ws[2 * t + 1]         = qp1;
    ws[2 * T + 2 * t] = kp0;  ws[2 * T + 2 * t + 1] = kp1;
    ws[4 * T + t]     = vop;
    ws[5 * T + t]     = varg;
    ws[6 * T + 2 * t] = qs0;  ws[6 * T + 2 * t + 1] = qs1;
    ws[8 * T + 2 * t] = ks0;  ws[8 * T + 2 * t + 1] = ks1;
    ws[10 * T + t]    = vs;
}

// ---------------------------------------------------------------------------
// Kernel 2: fused tiled scores + running argmax + gather, one head per
// blockIdx.y.  Each wave owns a 16-row Q tile.  K columns stream through a
// register->LDS double buffer; V_WMMA_F32_16X16X4_F32 produces the 16x16
// fp32 score tile (contraction dim padded 2 -> 4 with zeros).
//
// B operand trick: lanes 16..31 (which feed the zero-padded K=2,3 rows)
// read an 8-byte zero slot in LDS so the inner loop has a single
// unconditional ds_load_b64 per tile — no per-iteration exec masking.
// ---------------------------------------------------------------------------
__global__ __launch_bounds__(256) void argmax_fused(
    const float* __restrict__ ws, float* __restrict__ out, int T)
{
    __shared__ float sK[JCHUNK * 2 + 2];   // +2 floats: zero slot at JCHUNK*2

    const float* Q2; const float* K2; const float* V0; const float* V1;
    float* O0; float* O1;
    if (blockIdx.y == 0) {      // program-fetch head
        Q2 = ws;             K2 = ws + 2 * T;
        V0 = ws + 4 * T;     V1 = ws + 5 * T;
        O0 = out;            O1 = out + T;
    } else {                    // stack-fetch head
        Q2 = ws + 6 * T;     K2 = ws + 8 * T;
        V0 = ws + 10 * T;    V1 = nullptr;
        O0 = out + 2 * T;    O1 = nullptr;
    }

    const int tid  = threadIdx.x;
    const int lane = tid & 31;
    const int wave = tid >> 5;
    const int i0   = blockIdx.x * ROWS_PER_BLOCK + wave * 16;

    // A operand: this wave's 16 Q rows (lanes 16..31 feed padded K=2,3 -> 0)
    v2f a;
    if (lane < 16) {
        a.x = Q2[(i0 + lane) * 2];
        a.y = Q2[(i0 + lane) * 2 + 1];
    } else {
        a.x = 0.f; a.y = 0.f;
    }

    // Per-lane LDS walk for the B operand (set up once, loop is branchless)
    const float* bbase;
    int bstep;
    if (lane < 16) { bbase = sK + lane * 2;   bstep = 32; }  // 16 cols * 2 floats
    else           { bbase = sK + JCHUNK * 2; bstep = 0;  }  // zero slot

    if (tid == 0) { sK[JCHUNK * 2] = 0.f; sK[JCHUNK * 2 + 1] = 0.f; }

    float best[8];
    int   bidx[8];
#pragma unroll
    for (int v = 0; v < 8; ++v) { best[v] = -INFINITY; bidx[v] = 0x7fffffff; }

    const int ncol = lane & 15;   // column-within-tile this lane owns in C

    // register double-buffer: one float4 (2 K-columns' worth) per thread
    float4 stage = *(const float4*)(K2 + tid * 4);

    for (int jc = 0; jc < T; jc += JCHUNK) {
        *(float4*)(sK + tid * 4) = stage;
        __syncthreads();

        int jn = jc + JCHUNK;
        if (jn < T)  // issue next chunk's global load; overlaps WMMA sweep
            stage = *(const float4*)(K2 + jn * 2 + tid * 4);

        const float* bp = bbase;
#pragma unroll 4
        for (int jj = 0; jj < JCHUNK / 16; ++jj) {
            v2f b = *(const v2f*)bp;     // ds_load_b64, all 32 lanes
            bp += bstep;

            v8f c = {};
            c = __builtin_amdgcn_wmma_f32_16x16x4_f32(
                    /*neg_a=*/false, a, /*neg_b=*/false, b,
                    /*c_mod=*/(short)0, c, /*reuse_a=*/false, /*reuse_b=*/false);

            const int jcol = jc + jj * 16 + ncol;
#pragma unroll
            for (int v = 0; v < 8; ++v) {
                float cv = c[v];
                if (cv > best[v]) { best[v] = cv; bidx[v] = jcol; }
            }
        }
        __syncthreads();
    }

    // argmax reduce across the 16 lanes holding each row's columns;
    // tie-break toward the smaller index to match jnp.argmax.
#pragma unroll
    for (int v = 0; v < 8; ++v) {
        float m  = best[v];
        int   ix = bidx[v];
        for (int off = 8; off >= 1; off >>= 1) {
            float om = __shfl_xor(m, off, 16);
            int   oi = __shfl_xor(ix, off, 16);
            if (om > m || (om == m && oi < ix)) { m = om; ix = oi; }
        }
        if ((lane & 15) == 0) {
            int row = i0 + ((lane < 16) ? v : v + 8);
            O0[row] = V0[ix];
            if (O1) O1[row] = V1[ix];
        }
    }
}

extern "C" void kernel_launch(void* const* d_in, const int* in_sizes, int n_in,
                              void* d_out, int out_size, void* d_ws, size_t ws_size,
                              hipStream_t stream)
{
    const float* E    = (const float*)d_in[0];
    const float* WQp  = (const float*)d_in[1];
    const float* WKp  = (const float*)d_in[2];
    const float* WVop = (const float*)d_in[3];
    const float* WVarg= (const float*)d_in[4];
    const float* WQs  = (const float*)d_in[5];
    const float* WKs  = (const float*)d_in[6];
    const float* WVs  = (const float*)d_in[7];

    float* out = (float*)d_out;
    float* ws  = (float*)d_ws;

    const int T = in_sizes[0] / DIM;   // 8192

    proj_kernel<<<(T + 255) / 256, 256, 0, stream>>>(
        E, WQp, WKp, WVop, WVarg, WQs, WKs, WVs, ws, T);

    dim3 grid(T / ROWS_PER_BLOCK, 2);  // (64 row-blocks, 2 heads)
    argmax_fused<<<grid, 256, 0, stream>>>(ws, out, T);
}